// TrackMPNN_55808805044723
// MI455X (gfx1250) — compile-verified
//
#include <hip/hip_runtime.h>
#include <hip/hip_bf16.h>

#define NROWS 8192
#define NDET  4096

typedef float v2f __attribute__((ext_vector_type(2)));
typedef float v4f __attribute__((ext_vector_type(4)));
typedef float v8f __attribute__((ext_vector_type(8)));

// ---------------------------------------------------------------------------
// Kernel 1: PointNet branch + feature concat.  One 64-thread block per row.
// h0[row][0:64]  = x[row][0:64]
// h0[row][64:128]= maxpool_p( bn3(conv3( relu(bn2(conv2( relu(bn1(conv1(ch)))))))) )
// ---------------------------------------------------------------------------
__global__ void pointnet_kernel(
    const float* __restrict__ x,
    const float* __restrict__ w1, const float* __restrict__ b1,
    const float* __restrict__ g1, const float* __restrict__ be1,
    const float* __restrict__ m1, const float* __restrict__ v1,
    const float* __restrict__ w2, const float* __restrict__ b2,
    const float* __restrict__ g2, const float* __restrict__ be2,
    const float* __restrict__ m2, const float* __restrict__ v2,
    const float* __restrict__ w3, const float* __restrict__ b3,
    const float* __restrict__ g3, const float* __restrict__ be3,
    const float* __restrict__ m3, const float* __restrict__ v3,
    float* __restrict__ h0)
{
    __shared__ float s1[16][5];
    __shared__ float s2[32][5];
    const int row = blockIdx.x;
    const int t   = threadIdx.x;        // 64 threads
    const float* xr = x + (size_t)row * 74;

    // pass-through features
    h0[(size_t)row * 128 + t] = xr[t];

    // stage 1: 3->16 (third input channel is the zero pad)
    if (t < 16) {
        const float wa = w1[0 * 16 + t];
        const float wb = w1[1 * 16 + t];
        const float sc = g1[t] * rsqrtf(v1[t] + 1e-5f);
        const float sh = be1[t] - m1[t] * sc;
        const float bb = b1[t];
        for (int p = 0; p < 5; ++p) {
            float v = xr[64 + p] * wa + xr[69 + p] * wb + bb;
            s1[t][p] = fmaxf(v * sc + sh, 0.0f);
        }
    }
    __syncthreads();

    // stage 2: 16->32
    if (t < 32) {
        const float sc = g2[t] * rsqrtf(v2[t] + 1e-5f);
        const float sh = be2[t] - m2[t] * sc;
        const float bb = b2[t];
        for (int p = 0; p < 5; ++p) {
            float acc = bb;
            for (int c = 0; c < 16; ++c) acc += s1[c][p] * w2[c * 32 + t];
            s2[t][p] = fmaxf(acc * sc + sh, 0.0f);
        }
    }
    __syncthreads();

    // stage 3: 32->64, no relu, max-pool over the 5 points
    {
        const float sc = g3[t] * rsqrtf(v3[t] + 1e-5f);
        const float sh = be3[t] - m3[t] * sc;
        const float bb = b3[t];
        float mx = -3.4e38f;
        for (int p = 0; p < 5; ++p) {
            float acc = bb;
            for (int c = 0; c < 32; ++c) acc += s2[c][p] * w3[c * 64 + t];
            mx = fmaxf(mx, acc * sc + sh);
        }
        h0[(size_t)row * 128 + 64 + t] = mx;
    }
}

// ---------------------------------------------------------------------------
// Kernel 2: one-pass sparsification of the dense adjacencies.
// Row i < NDET comes from node_adj, otherwise from edge_adj (the other one is
// all-zero there and the where() picks exactly this branch).  <=9 nnz per row.
// One wave per row; non-temporal b128 streaming loads (256 MB read exactly
// once -> ~11 us @ 23.3 TB/s; TH=NT keeps the single-use stream from rinsing
// the L2 lines we actually want resident: h, weights, CSR arrays).
// ---------------------------------------------------------------------------
__global__ void sparsify_kernel(const float* __restrict__ node_adj,
                                const float* __restrict__ edge_adj,
                                int*   __restrict__ sp_cols,
                                float* __restrict__ sp_vals,
                                int*   __restrict__ sp_cnt)
{
    __shared__ int cnt[8];
    const int wid  = threadIdx.x >> 5;
    const int lane = threadIdx.x & 31;
    const int row  = blockIdx.x * 8 + wid;
    if (lane == 0) cnt[wid] = 0;
    __syncthreads();

    const float* src = (row < NDET ? node_adj : edge_adj) + (size_t)row * NROWS;
    const v4f* src4 = (const v4f*)src;

    for (int jj = lane; jj < NROWS / 4; jj += 32) {
        v4f v = __builtin_nontemporal_load(&src4[jj]);   // global_load_b128 th:NT
        const int cbase = jj * 4;
        if (v.x != 0.0f) { int s = atomicAdd(&cnt[wid], 1); if (s < 16) { sp_cols[row*16+s] = cbase+0; sp_vals[row*16+s] = v.x; } }
        if (v.y != 0.0f) { int s = atomicAdd(&cnt[wid], 1); if (s < 16) { sp_cols[row*16+s] = cbase+1; sp_vals[row*16+s] = v.y; } }
        if (v.z != 0.0f) { int s = atomicAdd(&cnt[wid], 1); if (s < 16) { sp_cols[row*16+s] = cbase+2; sp_vals[row*16+s] = v.z; } }
        if (v.w != 0.0f) { int s = atomicAdd(&cnt[wid], 1); if (s < 16) { sp_cols[row*16+s] = cbase+3; sp_vals[row*16+s] = v.w; } }
    }
    __syncthreads();
    if (lane == 0) sp_cnt[row] = min(cnt[wid], 16);
}

// ---------------------------------------------------------------------------
// Kernel 3: SpMM + concat.  z[row] = [ h[row], sum_k val_k * h[col_k] ].
// One block (C threads) per row; h (<=4MB) stays L2-resident.
// ---------------------------------------------------------------------------
__global__ void spmm_concat_kernel(const float* __restrict__ h, int C,
                                   const int*   __restrict__ sp_cols,
                                   const float* __restrict__ sp_vals,
                                   const int*   __restrict__ sp_cnt,
                                   float* __restrict__ z)
{
    __shared__ int   scols[16];
    __shared__ float svals[16];
    __shared__ int   scnt;
    const int row = blockIdx.x;
    const int t   = threadIdx.x;      // blockDim.x == C
    if (t < 16) { scols[t] = sp_cols[row * 16 + t]; svals[t] = sp_vals[row * 16 + t]; }
    if (t == 0) scnt = sp_cnt[row];
    __syncthreads();

    const float self = h[(size_t)row * C + t];
    float acc = 0.0f;
    const int cnt = scnt;
    for (int k = 0; k < cnt; ++k)
        acc += svals[k] * h[(size_t)scols[k] * C + t];

    z[(size_t)row * (2 * C) + t]     = self;
    z[(size_t)row * (2 * C) + C + t] = acc;
}

// ---------------------------------------------------------------------------
// Kernel 4: fused dual-branch linear layer via V_WMMA_F32_16X16X4_F32.
// out[8192, 64] = relu( z[8192, K] @ (wn|we)[K, 64] + (bn|be) ), branch chosen
// per 16-row tile (node rows < 4096, edge rows >= 4096; 4096 % 16 == 0).
// One wave computes one 16x64 tile (4 accumulators), 4 waves per block.
// A lane layout (16x4 f32):  row = lane%16, K-pair = 2*(lane/16)
// B lane layout (4x16 f32):  col = lane%16, K-pair = 2*(lane/16)
// C/D layout: vgpr r -> M = r + 8*(lane/16), N = lane%16
// ---------------------------------------------------------------------------
__global__ void fgc_gemm_kernel(const float* __restrict__ z, int K,
                                const float* __restrict__ wn,
                                const float* __restrict__ we,
                                const float* __restrict__ bn,
                                const float* __restrict__ be,
                                float* __restrict__ out, int relu)
{
    const int wid   = threadIdx.x >> 5;
    const int lane  = threadIdx.x & 31;
    const int mtile = blockIdx.x * 4 + wid;          // 512 tiles of 16 rows
    const int lm    = lane & 15;
    const int khalf = (lane >> 4) * 2;               // 0 or 2

    const bool isNode = (mtile * 16) < NDET;
    const float* __restrict__ W = isNode ? wn : we;  // [K, 64] row-major
    const float* __restrict__ B = isNode ? bn : be;  // [64]

    v8f acc0 = {}, acc1 = {}, acc2 = {}, acc3 = {};
    const float* __restrict__ zrow = z + (size_t)(mtile * 16 + lm) * K;

    for (int k0 = 0; k0 < K; k0 += 4) {
        const int k = k0 + khalf;
        v2f a;
        a.x = zrow[k];
        a.y = zrow[k + 1];
        const float* Wk = W + (size_t)k * 64;
        v2f b0, b1, b2, b3;
        b0.x = Wk[lm];          b0.y = Wk[64 + lm];
        b1.x = Wk[16 + lm];     b1.y = Wk[64 + 16 + lm];
        b2.x = Wk[32 + lm];     b2.y = Wk[64 + 32 + lm];
        b3.x = Wk[48 + lm];     b3.y = Wk[64 + 48 + lm];
        acc0 = __builtin_amdgcn_wmma_f32_16x16x4_f32(false, a, false, b0, (short)0, acc0, false, false);
        acc1 = __builtin_amdgcn_wmma_f32_16x16x4_f32(false, a, false, b1, (short)0, acc1, false, false);
        acc2 = __builtin_amdgcn_wmma_f32_16x16x4_f32(false, a, false, b2, (short)0, acc2, false, false);
        acc3 = __builtin_amdgcn_wmma_f32_16x16x4_f32(false, a, false, b3, (short)0, acc3, false, false);
    }

    const int mbase = mtile * 16 + (lane >> 4) * 8;
    for (int r = 0; r < 8; ++r) {
        float* orow = out + (size_t)(mbase + r) * 64;
        float v0 = acc0[r] + B[lm];
        float v1 = acc1[r] + B[16 + lm];
        float v2 = acc2[r] + B[32 + lm];
        float v3 = acc3[r] + B[48 + lm];
        if (relu) {
            v0 = fmaxf(v0, 0.0f); v1 = fmaxf(v1, 0.0f);
            v2 = fmaxf(v2, 0.0f); v3 = fmaxf(v3, 0.0f);
        }
        orow[lm]      = v0;
        orow[16 + lm] = v1;
        orow[32 + lm] = v2;
        orow[48 + lm] = v3;
    }
}

// ---------------------------------------------------------------------------
// Kernel 5: final 128->1 layer + sigmoid.  One wave per row.
// ---------------------------------------------------------------------------
__global__ void fgc_final_kernel(const float* __restrict__ z,   // [N, 128]
                                 const float* __restrict__ wn,  // [128, 1]
                                 const float* __restrict__ we,
                                 const float* __restrict__ bn,  // [1]
                                 const float* __restrict__ be,
                                 float* __restrict__ out)
{
    const int wid  = threadIdx.x >> 5;
    const int lane = threadIdx.x & 31;
    const int row  = blockIdx.x * 4 + wid;
    const bool isNode = row < NDET;
    const float* w = isNode ? wn : we;
    const float bias = isNode ? bn[0] : be[0];
    const float* zr = z + (size_t)row * 128;

    float acc = 0.0f;
    for (int j = lane; j < 128; j += 32) acc += zr[j] * w[j];
    for (int off = 16; off > 0; off >>= 1) acc += __shfl_xor(acc, off, 32);
    if (lane == 0) {
        float r = 1.0f / (1.0f + __expf(-(acc + bias)));
        __builtin_nontemporal_store(r, &out[row]);       // written once, never re-read
    }
}

// ---------------------------------------------------------------------------
extern "C" void kernel_launch(void* const* d_in, const int* in_sizes, int n_in,
                              void* d_out, int out_size, void* d_ws, size_t ws_size,
                              hipStream_t stream) {
    (void)in_sizes; (void)n_in; (void)out_size; (void)ws_size;

    const float* x        = (const float*)d_in[0];
    const float* node_adj = (const float*)d_in[1];
    const float* edge_adj = (const float*)d_in[2];
    // pn layer i (1-based): base 3 + (i-1)*6 -> w, b, g, be, m, v
    const float* pn[18];
    for (int i = 0; i < 18; ++i) pn[i] = (const float*)d_in[3 + i];
    // gc layer i (1-based): base 21 + (i-1)*4 -> wn, we, bn, be
    const float* gc1_wn = (const float*)d_in[21];
    const float* gc1_we = (const float*)d_in[22];
    const float* gc1_bn = (const float*)d_in[23];
    const float* gc1_be = (const float*)d_in[24];
    const float* gc2_wn = (const float*)d_in[25];
    const float* gc2_we = (const float*)d_in[26];
    const float* gc2_bn = (const float*)d_in[27];
    const float* gc2_be = (const float*)d_in[28];
    const float* gc3_wn = (const float*)d_in[29];
    const float* gc3_we = (const float*)d_in[30];
    const float* gc3_bn = (const float*)d_in[31];
    const float* gc3_be = (const float*)d_in[32];

    char* ws = (char*)d_ws;
    float* H0      = (float*)(ws);                       // 8192*128 f32 (4 MB)
    float* Z       = (float*)(ws + 4194304);             // 8192*256 f32 (8 MB)
    float* H1      = (float*)(ws + 12582912);            // 8192*64  f32 (2 MB)
    int*   sp_cols = (int*)  (ws + 14680064);            // 8192*16 i32
    float* sp_vals = (float*)(ws + 15204352);            // 8192*16 f32
    int*   sp_cnt  = (int*)  (ws + 15728640);            // 8192 i32

    // 0. PointNet branch + feature concat -> H0 [8192,128]
    pointnet_kernel<<<NROWS, 64, 0, stream>>>(
        x,
        pn[0],  pn[1],  pn[2],  pn[3],  pn[4],  pn[5],
        pn[6],  pn[7],  pn[8],  pn[9],  pn[10], pn[11],
        pn[12], pn[13], pn[14], pn[15], pn[16], pn[17],
        H0);

    // 1. One streaming (non-temporal) pass over both 256 MB adjacencies -> CSR-16
    sparsify_kernel<<<NROWS / 8, 256, 0, stream>>>(node_adj, edge_adj,
                                                   sp_cols, sp_vals, sp_cnt);

    // 2. gc1: SpMM+concat (C=128 -> K=256), WMMA GEMM -> H1 [8192,64], relu
    spmm_concat_kernel<<<NROWS, 128, 0, stream>>>(H0, 128, sp_cols, sp_vals, sp_cnt, Z);
    fgc_gemm_kernel<<<NROWS / 64, 128, 0, stream>>>(Z, 256, gc1_wn, gc1_we, gc1_bn, gc1_be, H1, 1);

    // 3. gc2: SpMM+concat (C=64 -> K=128), WMMA GEMM -> H0 (reused), relu
    spmm_concat_kernel<<<NROWS, 64, 0, stream>>>(H1, 64, sp_cols, sp_vals, sp_cnt, Z);
    fgc_gemm_kernel<<<NROWS / 64, 128, 0, stream>>>(Z, 128, gc2_wn, gc2_we, gc2_bn, gc2_be, H0, 1);

    // 4. gc3: SpMM+concat (C=64 -> K=128), 128->1 dot + sigmoid -> d_out
    spmm_concat_kernel<<<NROWS, 64, 0, stream>>>(H0, 64, sp_cols, sp_vals, sp_cnt, Z);
    fgc_final_kernel<<<NROWS / 4, 128, 0, stream>>>(Z, gc3_wn, gc3_we, gc3_bn, gc3_be,
                                                    (float*)d_out);
}